// NaiveCollider_58935541236175
// MI455X (gfx1250) — compile-verified
//
#include <hip/hip_runtime.h>
#include <stdint.h>

#define TS 256
#define EPSF 1e-9f

// ---- gfx1250 async global->LDS staging (ASYNCcnt path, CDNA5 ISA) ----
__device__ __forceinline__ unsigned lds_off_of(const void* p) {
  // generic pointer to LDS: low 32 bits are the LDS offset
  return (unsigned)(uintptr_t)p;
}
__device__ __forceinline__ void async_lds_b64(unsigned lds_off, const void* g) {
  asm volatile("global_load_async_to_lds_b64 %0, %1, off"
               :: "v"(lds_off), "v"(g) : "memory");
}
__device__ __forceinline__ void async_lds_b32(unsigned lds_off, const void* g) {
  asm volatile("global_load_async_to_lds_b32 %0, %1, off"
               :: "v"(lds_off), "v"(g) : "memory");
}
__device__ __forceinline__ void async_wait0() {
  asm volatile("s_wait_asynccnt 0" ::: "memory");
}

// One v_rsq_f32 (TRANS, ~1 ulp) replaces the precise sqrt+div fixup
// sequences (~15 VALU ops) in the inner loop; co-executes with VALU.
__device__ __forceinline__ float fast_rsq(float x) {
  return __builtin_amdgcn_rsqf(x);
}

// =================== circles: corr_cc + corr_c ===================
__global__ void collider_circles_kernel(const float* __restrict__ cpos,
                                        const float* __restrict__ crad,
                                        const float* __restrict__ apos,
                                        const float* __restrict__ ahalf,
                                        float* __restrict__ outp,
                                        int Nc, int Na, int S, int final_write) {
  __shared__ __align__(16) float sh_p[TS * 2];
  __shared__ __align__(16) float sh_r[TS];
  __shared__ __align__(16) float sh_h[TS * 2];

  const int tid = threadIdx.x;
  const int i = blockIdx.x * TS + tid;
  const int s = blockIdx.y;

  const float xi = cpos[2 * i];
  const float yi = cpos[2 * i + 1];
  const float ri = crad[i];
  float cx = 0.f, cy = 0.f;

  // ---- circle vs circle (self excluded automatically by d2<=EPS) ----
  const int ntc = Nc / TS;
  for (int t = s; t < ntc; t += S) {
    const int j = t * TS + tid;
    async_lds_b64(lds_off_of(&sh_p[2 * tid]), &cpos[2 * j]);
    async_lds_b32(lds_off_of(&sh_r[tid]), &crad[j]);
    async_wait0();
    __syncthreads();
#pragma unroll 8
    for (int k = 0; k < TS; ++k) {
      float dx = xi - sh_p[2 * k];
      float dy = yi - sh_p[2 * k + 1];
      float d2 = dx * dx + dy * dy;
      float d2c = d2 > EPSF ? d2 : 1.0f;
      float inv = fast_rsq(d2c);       // 1/dist
      float dist = d2c * inv;          // sqrt(d2c)
      float pen = ri + sh_r[k] - dist;
      float w = ((d2 > EPSF) && (pen > 0.f)) ? (0.5f * pen * inv) : 0.f;
      cx += w * dx;
      cy += w * dy;
    }
    __syncthreads();
  }

  // ---- circle vs aabb (positive push on circle) ----
  const int nta = Na / TS;
  for (int t = s; t < nta; t += S) {
    const int j = t * TS + tid;
    async_lds_b64(lds_off_of(&sh_p[2 * tid]), &apos[2 * j]);
    async_lds_b64(lds_off_of(&sh_h[2 * tid]), &ahalf[2 * j]);
    async_wait0();
    __syncthreads();
#pragma unroll 8
    for (int k = 0; k < TS; ++k) {
      float relx = xi - sh_p[2 * k];
      float rely = yi - sh_p[2 * k + 1];
      float hx = sh_h[2 * k];
      float hy = sh_h[2 * k + 1];
      float clx = fminf(fmaxf(relx, -hx), hx);
      float cly = fminf(fmaxf(rely, -hy), hy);
      float dx = relx - clx;
      float dy = rely - cly;
      float dd2 = dx * dx + dy * dy;
      float dd2c = dd2 > EPSF ? dd2 : 1.0f;
      float inv = fast_rsq(dd2c);
      float ddist = dd2c * inv;
      float pen = ri - ddist;
      float w = ((dd2 > EPSF) && (pen > 0.f)) ? (0.5f * pen * inv) : 0.f;
      cx += w * dx;
      cy += w * dy;
    }
    __syncthreads();
  }

  if (final_write) {
    outp[2 * i] = xi + cx;
    outp[2 * i + 1] = yi + cy;
  } else {
    outp[(size_t)(s * Nc + i) * 2] = cx;
    outp[(size_t)(s * Nc + i) * 2 + 1] = cy;
  }
}

// =================== aabbs: corr_aa + corr_a ===================
__global__ void collider_aabbs_kernel(const float* __restrict__ cpos,
                                      const float* __restrict__ crad,
                                      const float* __restrict__ apos,
                                      const float* __restrict__ ahalf,
                                      float* __restrict__ outp,
                                      int Nc, int Na, int S, int final_write) {
  __shared__ __align__(16) float sh_p[TS * 2];
  __shared__ __align__(16) float sh_r[TS];
  __shared__ __align__(16) float sh_h[TS * 2];

  const int tid = threadIdx.x;
  const int i = blockIdx.x * TS + tid;
  const int s = blockIdx.y;

  const float xi = apos[2 * i];
  const float yi = apos[2 * i + 1];
  const float hxi = ahalf[2 * i];
  const float hyi = ahalf[2 * i + 1];
  float cx = 0.f, cy = 0.f;

  // ---- aabb vs aabb (must exclude self by index) ----
  const int nta = Na / TS;
  for (int t = s; t < nta; t += S) {
    const int j = t * TS + tid;
    async_lds_b64(lds_off_of(&sh_p[2 * tid]), &apos[2 * j]);
    async_lds_b64(lds_off_of(&sh_h[2 * tid]), &ahalf[2 * j]);
    async_wait0();
    __syncthreads();
    const int jbase = t * TS;
#pragma unroll 8
    for (int k = 0; k < TS; ++k) {
      float dax = xi - sh_p[2 * k];
      float day = yi - sh_p[2 * k + 1];
      float ovx = hxi + sh_h[2 * k] - fabsf(dax);
      float ovy = hyi + sh_h[2 * k + 1] - fabsf(day);
      bool hit = ((jbase + k) != i) && (ovx > 0.f) && (ovy > 0.f);
      float sgx = (dax >= 0.f) ? 1.f : -1.f;
      float sgy = (day >= 0.f) ? 1.f : -1.f;
      bool use_x = (ovx <= ovy);
      float px = use_x ? (0.5f * ovx * sgx) : 0.f;
      float py = use_x ? 0.f : (0.5f * ovy * sgy);
      cx += hit ? px : 0.f;
      cy += hit ? py : 0.f;
    }
    __syncthreads();
  }

  // ---- circle vs aabb (equal-and-opposite on box) ----
  const int ntc = Nc / TS;
  for (int t = s; t < ntc; t += S) {
    const int j = t * TS + tid;
    async_lds_b64(lds_off_of(&sh_p[2 * tid]), &cpos[2 * j]);
    async_lds_b32(lds_off_of(&sh_r[tid]), &crad[j]);
    async_wait0();
    __syncthreads();
#pragma unroll 8
    for (int k = 0; k < TS; ++k) {
      float relx = sh_p[2 * k] - xi;   // circle - aabb, same as reference
      float rely = sh_p[2 * k + 1] - yi;
      float clx = fminf(fmaxf(relx, -hxi), hxi);
      float cly = fminf(fmaxf(rely, -hyi), hyi);
      float dx = relx - clx;
      float dy = rely - cly;
      float dd2 = dx * dx + dy * dy;
      float dd2c = dd2 > EPSF ? dd2 : 1.0f;
      float inv = fast_rsq(dd2c);
      float ddist = dd2c * inv;
      float pen = sh_r[k] - ddist;
      float w = ((dd2 > EPSF) && (pen > 0.f)) ? (0.5f * pen * inv) : 0.f;
      cx -= w * dx;                    // corr_a = -sum(push_ca)
      cy -= w * dy;
    }
    __syncthreads();
  }

  if (final_write) {
    outp[2 * i] = xi + cx;            // outp pre-offset by caller
    outp[2 * i + 1] = yi + cy;
  } else {
    outp[(size_t)(s * Na + i) * 2] = cx;
    outp[(size_t)(s * Na + i) * 2 + 1] = cy;
  }
}

// =================== slice reduction ===================
__global__ void collider_reduce_kernel(const float* __restrict__ cpos,
                                       const float* __restrict__ apos,
                                       const float* __restrict__ wsc,
                                       const float* __restrict__ wsa,
                                       float* __restrict__ outp,
                                       int Nc, int Na, int S) {
  const int idx = blockIdx.x * blockDim.x + threadIdx.x;
  if (idx < Nc) {
    float x = cpos[2 * idx], y = cpos[2 * idx + 1];
    for (int s = 0; s < S; ++s) {
      x += wsc[(size_t)(s * Nc + idx) * 2];
      y += wsc[(size_t)(s * Nc + idx) * 2 + 1];
    }
    outp[2 * idx] = x;
    outp[2 * idx + 1] = y;
  } else if (idx < Nc + Na) {
    const int a = idx - Nc;
    float x = apos[2 * a], y = apos[2 * a + 1];
    for (int s = 0; s < S; ++s) {
      x += wsa[(size_t)(s * Na + a) * 2];
      y += wsa[(size_t)(s * Na + a) * 2 + 1];
    }
    outp[2 * idx] = x;
    outp[2 * idx + 1] = y;
  }
}

extern "C" void kernel_launch(void* const* d_in, const int* in_sizes, int n_in,
                              void* d_out, int out_size, void* d_ws, size_t ws_size,
                              hipStream_t stream) {
  const float* cpos  = (const float*)d_in[0];
  const float* crad  = (const float*)d_in[1];
  const float* apos  = (const float*)d_in[2];
  const float* ahalf = (const float*)d_in[3];
  float* out = (float*)d_out;

  const int Nc = in_sizes[1];       // 4096
  const int Na = in_sizes[2] / 2;   // 2048
  const int S = 8;                  // j-slices for occupancy (192 -> 1536+ waves)

  const size_t need = (size_t)S * (size_t)(Nc + Na) * 2 * sizeof(float);
  if (ws_size >= need) {
    float* wsc = (float*)d_ws;                      // S * Nc * 2 floats
    float* wsa = wsc + (size_t)S * Nc * 2;          // S * Na * 2 floats
    collider_circles_kernel<<<dim3(Nc / TS, S), TS, 0, stream>>>(
        cpos, crad, apos, ahalf, wsc, Nc, Na, S, 0);
    collider_aabbs_kernel<<<dim3(Na / TS, S), TS, 0, stream>>>(
        cpos, crad, apos, ahalf, wsa, Nc, Na, S, 0);
    collider_reduce_kernel<<<(Nc + Na) / TS, TS, 0, stream>>>(
        cpos, apos, wsc, wsa, out, Nc, Na, S);
  } else {
    // fallback: direct single-slice path, no workspace
    collider_circles_kernel<<<dim3(Nc / TS, 1), TS, 0, stream>>>(
        cpos, crad, apos, ahalf, out, Nc, Na, 1, 1);
    collider_aabbs_kernel<<<dim3(Na / TS, 1), TS, 0, stream>>>(
        cpos, crad, apos, ahalf, out + (size_t)Nc * 2, Nc, Na, 1, 1);
  }
}